// Net_52123723104303
// MI455X (gfx1250) — compile-verified
//
#include <hip/hip_runtime.h>
#include <cmath>

#define IN_DIM 512
#define HID    64
#define NCLS   16

typedef __attribute__((ext_vector_type(2))) float v2f;
typedef __attribute__((ext_vector_type(8))) float v8f;

#define HAVE_WMMA_F32 __has_builtin(__builtin_amdgcn_wmma_f32_16x16x4_f32)

#if __has_builtin(__builtin_amdgcn_global_load_async_to_lds_b128) && \
    __has_builtin(__builtin_amdgcn_s_wait_asynccnt)
#define USE_ASYNC 1
// 16-byte vector type the async builtin expects, in global / LDS address spaces.
typedef int v4i_vs __attribute__((vector_size(16)));
typedef v4i_vs __attribute__((address_space(1)))* async_gptr;
typedef v4i_vs __attribute__((address_space(3)))* async_lptr;
#else
#define USE_ASYNC 0
#endif

// ---------------------------------------------------------------------------
// Degree / normalization kernels
// ---------------------------------------------------------------------------
__global__ __launch_bounds__(256) void fill_deg(float* deg, int n) {
    int i = blockIdx.x * blockDim.x + threadIdx.x;
    if (i < n) deg[i] = 1.0f;                       // self-loop weight
}

__global__ __launch_bounds__(256) void accum_deg(const int* __restrict__ dst,
                                                 const float* __restrict__ w,
                                                 float* __restrict__ deg, int e) {
    int i = blockIdx.x * blockDim.x + threadIdx.x;
    if (i < e) atomicAdd(&deg[dst[i]], w[i]);
}

__global__ __launch_bounds__(256) void finish_deg(float* deg, int n) {
    int i = blockIdx.x * blockDim.x + threadIdx.x;
    if (i < n) deg[i] = rsqrtf(deg[i]);             // deg >= 1 always
}

// ---------------------------------------------------------------------------
// GEMM: C[M,NDIM] = (relu?)A[M,KDIM] @ B[KDIM,NDIM] using V_WMMA_F32_16X16X4_F32.
// One wave owns a 16-row tile and all NDIM columns (NDIM/16 accumulators),
// so each row of A streams from HBM exactly once. B is staged into LDS once
// per workgroup via GLOBAL_LOAD_ASYNC_TO_LDS_B128 (ASYNCcnt-tracked), so the
// per-k-step B-operand reads come from LDS instead of hammering L2.
// ---------------------------------------------------------------------------
#if HAVE_WMMA_F32
template <int KDIM, int NDIM, bool RELU_IN>
__global__ __launch_bounds__(256) void gemm_kern(const float* __restrict__ A,
                                                 const float* __restrict__ Bm,
                                                 float* __restrict__ C, int M) {
    extern __shared__ float Blds[];                 // KDIM*NDIM floats

    // ---- stage B into LDS (all 8 waves participate, then barrier) ----
    constexpr int NV16 = KDIM * NDIM / 4;           // number of 16-byte chunks
#if USE_ASYNC
    for (int v = threadIdx.x; v < NV16; v += 256)
        __builtin_amdgcn_global_load_async_to_lds_b128(
            (async_gptr)(Bm + 4 * (size_t)v), (async_lptr)(&Blds[4 * v]), 0, 0);
    __builtin_amdgcn_s_wait_asynccnt(0);
#else
    for (int v = threadIdx.x; v < NV16; v += 256)
        *reinterpret_cast<float4*>(&Blds[4 * v]) =
            *reinterpret_cast<const float4*>(&Bm[4 * (size_t)v]);
#endif
    __syncthreads();

    const int wave = blockIdx.x * 8 + (threadIdx.x >> 5);
    const int lane = threadIdx.x & 31;
    if (wave * 16 >= M) return;                     // whole-wave exit: EXEC stays all-1s
    const int col = lane & 15;
    const int kg  = lane >> 4;                      // lanes 16-31 hold K+2 / K+3
    const int row = wave * 16 + col;

    constexpr int NT = NDIM / 16;
    v8f acc[NT] = {};

    const float* arow = A + (size_t)row * KDIM + 2 * kg;
    const float* bl   = Blds + (2 * kg) * NDIM + col;

    for (int k0 = 0; k0 < KDIM; k0 += 4) {
        float a0 = arow[0];
        float a1 = arow[1];
        if (RELU_IN) { a0 = fmaxf(a0, 0.0f); a1 = fmaxf(a1, 0.0f); }
        v2f a;
        a.x = a0;
        a.y = a1;
        arow += 4;
#pragma unroll
        for (int t = 0; t < NT; ++t) {
            v2f b;
            b.x = bl[t * 16];                       // K = k0 + 2*kg
            b.y = bl[t * 16 + NDIM];                // K = k0 + 2*kg + 1
            acc[t] = __builtin_amdgcn_wmma_f32_16x16x4_f32(
                false, a, false, b, (short)0, acc[t], false, false);
        }
        bl += 4 * NDIM;
    }

    // C/D layout: VGPR r -> row r (lanes 0-15) / row r+8 (lanes 16-31), N = lane%16
#pragma unroll
    for (int t = 0; t < NT; ++t)
#pragma unroll
        for (int r = 0; r < 8; ++r)
            C[(size_t)(wave * 16 + r + 8 * kg) * NDIM + t * 16 + col] = acc[t][r];
}
#else  // fallback keeps the file compiling; histogram (wmma==0) signals to switch builtin
template <int KDIM, int NDIM, bool RELU_IN>
__global__ __launch_bounds__(256) void gemm_kern(const float* __restrict__ A,
                                                 const float* __restrict__ Bm,
                                                 float* __restrict__ C, int M) {
    int i = blockIdx.x * blockDim.x + threadIdx.x;
    if (i >= M * NDIM) return;
    int m = i / NDIM, n = i % NDIM;
    float s = 0.0f;
    for (int k = 0; k < KDIM; ++k) {
        float a = A[(size_t)m * KDIM + k];
        if (RELU_IN) a = fmaxf(a, 0.0f);
        s += a * Bm[(size_t)k * NDIM + n];
    }
    C[i] = s;
}
#endif

// ---------------------------------------------------------------------------
// Bias broadcast init
// ---------------------------------------------------------------------------
__global__ __launch_bounds__(256) void init_bias(float* __restrict__ o,
                                                 const float* __restrict__ b,
                                                 int total, int mask) {
    int i = blockIdx.x * blockDim.x + threadIdx.x;
    if (i < total) o[i] = b[i & mask];
}

// ---------------------------------------------------------------------------
// Edge scatter, 64-wide features: 16 lanes per edge, float4 gather + 4 atomics.
// h1 (25.6 MB) and a1 both live in the 192 MB L2, so the random gather and the
// f32 atomics stay on-chip. Edges [E, E+N) are self-loops with weight 1.
// ---------------------------------------------------------------------------
__global__ __launch_bounds__(256) void scatter_h64(const int* __restrict__ src,
                                                   const int* __restrict__ dst,
                                                   const float* __restrict__ w,
                                                   const float* __restrict__ dis,
                                                   const float* __restrict__ h,
                                                   float* __restrict__ out,
                                                   int E, int N) {
    long long t = (long long)blockIdx.x * blockDim.x + threadIdx.x;
    int e = (int)(t >> 4);
    if (e >= E + N) return;
    int c = ((int)t & 15) * 4;
    int s, d;
    float wt;
    if (e < E) { s = src[e]; d = dst[e]; wt = w[e]; }
    else       { s = e - E;  d = s;      wt = 1.0f; }
    float nrm = dis[s] * wt * dis[d];
    const float4 hv = *reinterpret_cast<const float4*>(h + (size_t)s * HID + c);
    float* o = out + (size_t)d * HID + c;
    atomicAdd(o + 0, hv.x * nrm);
    atomicAdd(o + 1, hv.y * nrm);
    atomicAdd(o + 2, hv.z * nrm);
    atomicAdd(o + 3, hv.w * nrm);
}

// 16-wide features: 1 lane per class, 16 lanes per edge.
__global__ __launch_bounds__(256) void scatter_h16(const int* __restrict__ src,
                                                   const int* __restrict__ dst,
                                                   const float* __restrict__ w,
                                                   const float* __restrict__ dis,
                                                   const float* __restrict__ h,
                                                   float* __restrict__ out,
                                                   int E, int N) {
    long long t = (long long)blockIdx.x * blockDim.x + threadIdx.x;
    int e = (int)(t >> 4);
    if (e >= E + N) return;
    int c = (int)t & 15;
    int s, d;
    float wt;
    if (e < E) { s = src[e]; d = dst[e]; wt = w[e]; }
    else       { s = e - E;  d = s;      wt = 1.0f; }
    float nrm = dis[s] * wt * dis[d];
    atomicAdd(&out[(size_t)d * NCLS + c], h[(size_t)s * NCLS + c] * nrm);
}

// ---------------------------------------------------------------------------
// In-place log-softmax over 16 classes, one thread per node.
// ---------------------------------------------------------------------------
__global__ __launch_bounds__(256) void logsoftmax16(float* io, int n) {
    int i = blockIdx.x * blockDim.x + threadIdx.x;
    if (i >= n) return;
    float* p = io + (size_t)i * NCLS;
    float v[NCLS];
    float mx = -INFINITY;
#pragma unroll
    for (int j = 0; j < NCLS; ++j) { v[j] = p[j]; mx = fmaxf(mx, v[j]); }
    float s = 0.0f;
#pragma unroll
    for (int j = 0; j < NCLS; ++j) s += expf(v[j] - mx);
    float lse = mx + logf(s);
#pragma unroll
    for (int j = 0; j < NCLS; ++j) p[j] = v[j] - lse;
}

// ---------------------------------------------------------------------------
// Launch
// ---------------------------------------------------------------------------
extern "C" void kernel_launch(void* const* d_in, const int* in_sizes, int n_in,
                              void* d_out, int out_size, void* d_ws, size_t ws_size,
                              hipStream_t stream) {
    const float* x  = (const float*)d_in[0];
    const int*   ei = (const int*)d_in[1];
    const float* ew = (const float*)d_in[2];
    const float* W1 = (const float*)d_in[3];
    const float* b1 = (const float*)d_in[4];
    const float* W2 = (const float*)d_in[5];
    const float* b2 = (const float*)d_in[6];

    const int N = in_sizes[0] / IN_DIM;
    const int E = in_sizes[1] / 2;
    const int* srcp = ei;
    const int* dstp = ei + E;

    float* ws  = (float*)d_ws;
    float* dis = ws;                         // N
    float* h1  = dis + (size_t)N;            // N*HID
    float* a1  = h1 + (size_t)N * HID;       // N*HID
    float* h2  = a1 + (size_t)N * HID;       // N*NCLS
    float* out = (float*)d_out;

    const int B = 256;
    auto blk = [](long long total, int b) { return (unsigned)((total + b - 1) / b); };

    // 1) symmetric normalization coefficients
    fill_deg<<<blk(N, B), B, 0, stream>>>(dis, N);
    accum_deg<<<blk(E, B), B, 0, stream>>>(dstp, ew, dis, E);
    finish_deg<<<blk(N, B), B, 0, stream>>>(dis, N);

    // 2) h1 = x @ W1
    {
        int tiles = (N + 15) / 16;
#if HAVE_WMMA_F32
        gemm_kern<IN_DIM, HID, false>
            <<<blk(tiles, 8), B, IN_DIM * HID * sizeof(float), stream>>>(x, W1, h1, N);
#else
        gemm_kern<IN_DIM, HID, false><<<blk((long long)N * HID, B), B, 0, stream>>>(x, W1, h1, N);
#endif
    }

    // 3) a1 = b1; a1[dst] += norm * h1[src] (edges + self-loops)
    init_bias<<<blk((long long)N * HID, B), B, 0, stream>>>(a1, b1, N * HID, HID - 1);
    scatter_h64<<<blk((long long)(E + N) * 16, B), B, 0, stream>>>(srcp, dstp, ew, dis, h1, a1, E, N);

    // 4) h2 = relu(a1) @ W2   (ReLU fused into A-operand load)
    {
        int tiles = (N + 15) / 16;
#if HAVE_WMMA_F32
        gemm_kern<HID, NCLS, true>
            <<<blk(tiles, 8), B, HID * NCLS * sizeof(float), stream>>>(a1, W2, h2, N);
#else
        gemm_kern<HID, NCLS, true><<<blk((long long)N * NCLS, B), B, 0, stream>>>(a1, W2, h2, N);
#endif
    }

    // 5) out = b2; out[dst] += norm * h2[src]; log_softmax in place
    init_bias<<<blk((long long)N * NCLS, B), B, 0, stream>>>(out, b2, N * NCLS, NCLS - 1);
    scatter_h16<<<blk((long long)(E + N) * 16, B), B, 0, stream>>>(srcp, dstp, ew, dis, h2, out, E, N);
    logsoftmax16<<<blk(N, B), B, 0, stream>>>(out, N);
}